// LimbRelations_45346264711202
// MI455X (gfx1250) — compile-verified
//
#include <hip/hip_runtime.h>

typedef __attribute__((ext_vector_type(2))) float v2f;
typedef __attribute__((ext_vector_type(8))) float v8f;

#define XCOLS   75
#define OUTCOLS 1355
#define WROW    26   // padded LDS row stride for W (24 cols + 2 pad -> no 24-stride bank conflicts)

// D = A(16x4) * B(4x16) + C(16x16), all fp32 — exact match to the fp32 reference.
static __device__ __forceinline__ v8f wmma4(v2f a, v2f b, v8f c) {
  // (neg_a, A, neg_b, B, c_mod, C, reuse_a, reuse_b)
  return __builtin_amdgcn_wmma_f32_16x16x4_f32(false, a, false, b, (short)0, c, false, false);
}

__global__ __launch_bounds__(256)
void limb_rel_kernel(const float* __restrict__ x, const float* __restrict__ W,
                     const float* __restrict__ b, float* __restrict__ out, int T)
{
  __shared__ float sW[128 * WROW];   // 13312 B, padded
  __shared__ float sB[128];

  const int tid  = threadIdx.x;
  const int lane = tid & 31;
  const int wave = tid >> 5;
  const int row  = lane & 15;          // M (or N) index within 16
  const int khi  = (lane >> 4) << 1;   // K offset 0 / 2 (A,B 32-bit fragment layout)
  const int moff = (lane >> 4) << 3;   // C/D: upper half-wave holds M+8

  // Stage W (row-major 128x24) into padded LDS; stage bias.
  for (int i = tid; i < 128 * 24; i += 256) {
    int r = i / 24;
    int c = i - r * 24;
    sW[r * WROW + c] = W[i];
  }
  if (tid < 128) sB[tid] = b[tid];
  __syncthreads();

  const long tile = (long)blockIdx.x * 8 + wave;   // one 16-frame tile per wave
  const long base = tile * 16;
  if (base >= T) return;                            // wave-uniform: EXEC stays all-1 for WMMA

  // ---- pass-through: out[:, 0:75] = x (coalesced) ----
  {
    const float* xin = x + base * XCOLS;
    float* o = out + base * (long)OUTCOLS;
    #pragma unroll
    for (int r = 0; r < 16; ++r) {
      const float* xr = xin + r * XCOLS;
      float* orow = o + r * (long)OUTCOLS;
      orow[lane]      = xr[lane];
      orow[lane + 32] = xr[lane + 32];
      if (lane < XCOLS - 64) orow[lane + 64] = xr[lane + 64];
    }
  }

  // ---- A fragments: 5 limbs x K=12 (3 ksteps of 4), ISA 16x4 f32 layout ----
  v2f afr[5][3];
  {
    const float* xr = x + (base + row) * XCOLS;
    #pragma unroll
    for (int lm = 0; lm < 5; ++lm) {
      #pragma unroll
      for (int s = 0; s < 3; ++s) {
        const float* p = xr + lm * 12 + s * 4 + khi;
        v2f a; a.x = p[0]; a.y = p[1];
        afr[lm][s] = a;
      }
    }
  }

  const int P1[10] = {0,0,0,0,1,1,1,2,2,3};   // np.triu_indices(5,1)
  const int P2[10] = {1,2,3,4,2,3,4,3,4,4};

  float* obase = out + base * (long)OUTCOLS + 75;

  #pragma unroll 1
  for (int nt = 0; nt < 8; ++nt) {            // 8 N-tiles of 16 output channels
    // B fragments: B[k][n] = W[n][k]; split W into W1 (cols 0..11) and W2 (cols 12..23)
    const float* wr = sW + (nt * 16 + row) * WROW;
    v2f b1[3], b2[3];
    #pragma unroll
    for (int s = 0; s < 3; ++s) {
      v2f t;
      t.x = wr[s * 4 + khi];      t.y = wr[s * 4 + khi + 1];      b1[s] = t;
      t.x = wr[12 + s * 4 + khi]; t.y = wr[12 + s * 4 + khi + 1]; b2[s] = t;
    }
    const float bias = sB[nt * 16 + row];

    // U1[l] = limb[l] @ W1^T ; U2[l] = limb[l] @ W2^T  (10 accumulators, 30 WMMAs)
    v8f u1[5], u2[5];
    #pragma unroll
    for (int lm = 0; lm < 5; ++lm) {
      v8f c1 = {};
      c1 = wmma4(afr[lm][0], b1[0], c1);
      c1 = wmma4(afr[lm][1], b1[1], c1);
      c1 = wmma4(afr[lm][2], b1[2], c1);
      u1[lm] = c1;
      v8f c2 = {};
      c2 = wmma4(afr[lm][0], b2[0], c2);
      c2 = wmma4(afr[lm][1], b2[1], c2);
      c2 = wmma4(afr[lm][2], b2[2], c2);
      u2[lm] = c2;
    }

    // pair = U1[i] + U2[j] + b, ReLU, coalesced 64B stores per half-wave
    #pragma unroll
    for (int p = 0; p < 10; ++p) {
      v8f d = u1[P1[p]] + u2[P2[p]];
      float* op = obase + p * 128 + nt * 16 + row;
      #pragma unroll
      for (int r = 0; r < 8; ++r) {
        float v = d[r] + bias;
        v = v > 0.0f ? v : 0.0f;
        op[(long)(moff + r) * OUTCOLS] = v;
      }
    }
  }
}

extern "C" void kernel_launch(void* const* d_in, const int* in_sizes, int n_in,
                              void* d_out, int out_size, void* d_ws, size_t ws_size,
                              hipStream_t stream) {
  const float* x = (const float*)d_in[0];
  const float* W = (const float*)d_in[1];
  const float* b = (const float*)d_in[2];
  float* out = (float*)d_out;
  const int T = in_sizes[0] / XCOLS;          // 262144
  const int tiles  = (T + 15) / 16;           // 16-frame tiles
  const int blocks = (tiles + 7) / 8;         // 8 waves (tiles) per 256-thread block
  limb_rel_kernel<<<blocks, 256, 0, stream>>>(x, W, b, out, T);
}